// DecoderBase_1365799600407
// MI455X (gfx1250) — compile-verified
//
#include <hip/hip_runtime.h>

// ---------------------------------------------------------------------------
// MI455X / gfx1250 implementation.
//
// Reference: 6x [conv3x3(64->16) + bias + relu] -> 2x nearest upsample ->
// 6-way broadcast product into voxel (1,16,64,64,32,16) f32 = 128 MB.
//
// Roofline: output write (128 MB) dominates: ~5.5 us at 23.3 TB/s HBM.
// Conv is ~50 MFLOP (noise) but is mapped onto V_WMMA_F32_16X16X4_F32 as an
// implicit GEMM (M=spatial, N=16 outch, K=64*9=576) to use the matrix path.
// Upsample is folded into indexing (>>1) so only half-res planes are stored.
// ---------------------------------------------------------------------------

typedef __attribute__((ext_vector_type(2))) float v2f;
typedef __attribute__((ext_vector_type(4))) float v4f;
typedef __attribute__((ext_vector_type(8))) float v8f;

// Half-res conv-output plane offsets (floats) inside d_ws, layout [H][W][16c]:
//   xy: 32x32  @ 0        (16384)
//   xz: 32x16  @ 16384    ( 8192)
//   yz: 32x16  @ 24576    ( 8192)
//   tx:  8x32  @ 32768    ( 4096)
//   ty:  8x32  @ 36864    ( 4096)
//   tz:  8x16  @ 40960    ( 2048)   total 43008 floats = 172 KB

// ===========================================================================
// Kernel 1: implicit-GEMM conv3x3 + bias + ReLU via V_WMMA_F32_16X16X4_F32.
// One wave (32 threads) per 16-spatial-position tile; 168 tiles total.
// A (16x4 f32): lane = M-row (lane&15); lanes>=16 carry K=2,3 (ISA layout).
// B (4x16 f32): lane = N-col (lane&15); lanes>=16 carry K-rows 2,3.
// D (16x16 f32): 8 VGPRs, M = v + 8*(lane>=16), N = lane&15.
// ===========================================================================
__global__ __launch_bounds__(32) void conv_wmma_kernel(
    const float* __restrict__ pxy, const float* __restrict__ pxz,
    const float* __restrict__ pyz, const float* __restrict__ ptx,
    const float* __restrict__ pty, const float* __restrict__ ptz,
    const float* __restrict__ Wt,  const float* __restrict__ Bias,
    float* __restrict__ ws)
{
    const int b = blockIdx.x;                // 0..167 (tile id, uniform)
    const float* P; int p, tile, Hp, Wp, wsh, ob;
    if      (b <  64) { p=0; tile=b;     P=pxy; Hp=32; Wp=32; wsh=5; ob=0;     }
    else if (b <  96) { p=1; tile=b-64;  P=pxz; Hp=32; Wp=16; wsh=4; ob=16384; }
    else if (b < 128) { p=2; tile=b-96;  P=pyz; Hp=32; Wp=16; wsh=4; ob=24576; }
    else if (b < 144) { p=3; tile=b-128; P=ptx; Hp=8;  Wp=32; wsh=5; ob=32768; }
    else if (b < 160) { p=4; tile=b-144; P=pty; Hp=8;  Wp=32; wsh=5; ob=36864; }
    else              { p=5; tile=b-160; P=ptz; Hp=8;  Wp=16; wsh=4; ob=40960; }
    const int HW = Hp * Wp;

    const int lane = threadIdx.x;
    const int mrow = lane & 15;          // M row this lane feeds for A
    const int khi  = lane >> 4;          // 0: K={0,1}, 1: K={2,3} within chunk
    const int n    = lane & 15;          // N col this lane feeds for B / D
    const int m    = tile * 16 + mrow;   // spatial position (all planes are
    const int h    = m >> wsh;           //  multiples of 16 positions)
    const int w    = m & (Wp - 1);

    v8f acc0 = {0,0,0,0,0,0,0,0};        // two accumulators to break the
    v8f acc1 = {0,0,0,0,0,0,0,0};        // serial WMMA D->C dependency chain

    const float* wbase = Wt + (size_t)(p * 16 + n) * 64 * 9; // OIHW weights

    for (int kh = 0; kh < 3; ++kh) {
        const int ih = h + kh - 1;
        for (int kw = 0; kw < 3; ++kw) {
            const int iw = w + kw - 1;
            const bool inb = ((unsigned)ih < (unsigned)Hp) &&
                             ((unsigned)iw < (unsigned)Wp);
            const float msk = inb ? 1.0f : 0.0f;       // SAME zero padding
            const float* src = P + (inb ? (ih * Wp + iw) : 0); // safe addr
            const float* wk  = wbase + kh * 3 + kw;
            #pragma unroll
            for (int c0 = 0; c0 < 64; c0 += 8) {
                {
                    const int c = c0 + 2 * khi;
                    v2f a;  a.x  = msk * src[c * HW];  a.y  = msk * src[(c + 1) * HW];
                    v2f bb; bb.x = wk[c * 9];          bb.y = wk[(c + 1) * 9];
                    acc0 = __builtin_amdgcn_wmma_f32_16x16x4_f32(
                        false, a, false, bb, (short)0, acc0, false, false);
                }
                {
                    const int c = c0 + 4 + 2 * khi;
                    v2f a;  a.x  = msk * src[c * HW];  a.y  = msk * src[(c + 1) * HW];
                    v2f bb; bb.x = wk[c * 9];          bb.y = wk[(c + 1) * 9];
                    acc1 = __builtin_amdgcn_wmma_f32_16x16x4_f32(
                        false, a, false, bb, (short)0, acc1, false, false);
                }
            }
        }
    }
    v8f acc = acc0 + acc1;

    const float bias = Bias[p * 16 + n];
    float* op = ws + ob;
    #pragma unroll
    for (int v = 0; v < 8; ++v) {
        const int mm  = tile * 16 + v + 8 * khi;   // D-matrix row -> spatial
        float val = acc[v] + bias;
        val = val > 0.0f ? val : 0.0f;             // ReLU
        op[mm * 16 + n] = val;                     // [h][w][c] layout
    }
}

// ===========================================================================
// Kernel 2: voxel product + implicit 2x upsample. Store-bandwidth bound.
// One block per (t,x): writes a contiguous 128 KB tile out[t][x][.][.][.].
// LDS: yz plane copy (32 KB) + precomputed ay = xy*ty*tx (2 KB) and
// bz = xz*tz (1 KB). Inner loop: 2 LDS b128 loads + 8 muls per b128 store.
// ===========================================================================
__global__ __launch_bounds__(256) void voxel_kernel(
    const float* __restrict__ ws, float* __restrict__ out)
{
    const int tb = blockIdx.x;         // 0..1023
    const int t  = tb >> 6;            // 0..15
    const int x  = tb & 63;            // 0..63
    const int t2 = t >> 1, x2 = x >> 1;
    const int tid = threadIdx.x;

    __shared__ __align__(16) float lds_ay[32 * 16];       //  2 KB
    __shared__ __align__(16) float lds_bz[16 * 16];       //  1 KB
    __shared__ __align__(16) float lds_yz[32 * 16 * 16];  // 32 KB

    const float* xy = ws;
    const float* xz = ws + 16384;
    const float* yz = ws + 24576;
    const float* tx = ws + 32768;
    const float* ty = ws + 36864;
    const float* tz = ws + 40960;

    // Stage the full half-res yz plane (2048 float4).
    #pragma unroll
    for (int j = 0; j < 8; ++j) {
        const int f = j * 256 + tid;
        ((v4f*)lds_yz)[f] = ((const v4f*)yz)[f];
    }
    // ay[y2][c] = xy[x2][y2][c] * ty[t2][y2][c] * tx[t2][x2][c]
    if (tid < 128) {
        const int c4 = tid & 3;
        const int y2 = tid >> 2;                         // 0..31
        v4f a = ((const v4f*)(xy + (x2 * 32 + y2) * 16))[c4];
        v4f bq = ((const v4f*)(ty + (t2 * 32 + y2) * 16))[c4];
        v4f c = ((const v4f*)(tx + (t2 * 32 + x2) * 16))[c4];
        ((v4f*)lds_ay)[y2 * 4 + c4] = a * bq * c;
    } else if (tid < 192) {
        // bz[z2][c] = xz[x2][z2][c] * tz[t2][z2][c]
        const int f  = tid - 128;
        const int c4 = f & 3;
        const int z2 = f >> 2;                           // 0..15
        v4f a = ((const v4f*)(xz + (x2 * 16 + z2) * 16))[c4];
        v4f bq = ((const v4f*)(tz + (t2 * 16 + z2) * 16))[c4];
        ((v4f*)lds_bz)[z2 * 4 + c4] = a * bq;
    }
    __syncthreads();

    // Each thread keeps (z, c4) fixed -> bz held in registers; stream y.
    const int c4   = tid & 3;            // float4 index within 16 channels
    const int z    = (tid >> 2) & 31;    // full-res z
    const int zh   = z >> 1;
    const int ybit = tid >> 7;           // threads 0-127: even y, 128-255: odd
    const v4f bz   = ((const v4f*)lds_bz)[zh * 4 + c4];

    float* obase = out + ((size_t)tb << 15);     // tb * 64*32*16 floats
    #pragma unroll 4
    for (int it = 0; it < 32; ++it) {
        const int y  = 2 * it + ybit;            // y>>1 == it
        const v4f ay  = ((const v4f*)lds_ay)[it * 4 + c4];
        const v4f vyz = ((const v4f*)lds_yz)[(it * 16 + zh) * 4 + c4];
        const v4f r = ay * bz * vyz;
        ((v4f*)(obase + (size_t)(y * 32 + z) * 16))[c4] = r;   // coalesced b128
    }
}

extern "C" void kernel_launch(void* const* d_in, const int* in_sizes, int n_in,
                              void* d_out, int out_size, void* d_ws, size_t ws_size,
                              hipStream_t stream) {
    (void)in_sizes; (void)n_in; (void)out_size; (void)ws_size;
    const float* pxy = (const float*)d_in[0];
    const float* pxz = (const float*)d_in[1];
    const float* pyz = (const float*)d_in[2];
    const float* ptx = (const float*)d_in[3];
    const float* pty = (const float*)d_in[4];
    const float* ptz = (const float*)d_in[5];
    const float* Wt  = (const float*)d_in[6];   // (6,16,64,3,3)
    const float* Bs  = (const float*)d_in[7];   // (6,16)
    float* ws  = (float*)d_ws;
    float* out = (float*)d_out;

    // 168 tiles of 16 spatial positions, one wave each.
    conv_wmma_kernel<<<168, 32, 0, stream>>>(pxy, pxz, pyz, ptx, pty, ptz,
                                             Wt, Bs, ws);
    // 1024 (t,x) tiles, 128 KB contiguous output each.
    voxel_kernel<<<1024, 256, 0, stream>>>(ws, out);
}